// MoE_27693949124969
// MI455X (gfx1250) — compile-verified
//
#include <hip/hip_runtime.h>

typedef __attribute__((ext_vector_type(2))) float v2f;
typedef __attribute__((ext_vector_type(8))) float v8f;

#define NTOK     16384            // B*S = 4*4096
#define NEXP     8
#define DDIM     1024
#define RT       32               // rows (token-expert pairs) per tile
#define MAXPAIR  (2 * NTOK)       // 32768 (top_k = 2 -> exactly 2 per token)
#define MAXTILES (MAXPAIR / RT + NEXP)   // 1032 worst-case padded tiles
#define PAIRCAP  (MAXTILES * RT)         // 33024
#define KC       256              // K chunk staged in LDS

// ---------------------------------------------------------------------------
// Kernel 1: initialize output + scratch (harness poisons buffers with 0xAA)
// ---------------------------------------------------------------------------
__global__ __launch_bounds__(256)
void moe_init(float* __restrict__ out, int* __restrict__ cnt,
              int* __restrict__ pairTok, float* __restrict__ pairW)
{
    const size_t nout = (size_t)NTOK * DDIM;
    size_t i0 = (size_t)blockIdx.x * blockDim.x + threadIdx.x;
    size_t stride = (size_t)gridDim.x * blockDim.x;
    for (size_t i = i0; i < nout; i += stride) out[i] = 0.0f;
    for (size_t i = i0; i < (size_t)PAIRCAP; i += stride) { pairTok[i] = -1; pairW[i] = 0.0f; }
    if (i0 < NEXP) cnt[i0] = 0;
}

// ---------------------------------------------------------------------------
// Kernel 2: gating -- one wave per token, top-2 raw logits as combine weights
// ---------------------------------------------------------------------------
__global__ __launch_bounds__(256)
void moe_gate(const float* __restrict__ x, const float* __restrict__ Wg,
              const float* __restrict__ bg, int* __restrict__ topIdx,
              float* __restrict__ topW, int* __restrict__ cnt)
{
    const int wave  = threadIdx.x >> 5;
    const int lane  = threadIdx.x & 31;
    const int token = blockIdx.x * 8 + wave;
    const float* xr = x + (size_t)token * DDIM;

    float acc[8];
#pragma unroll
    for (int e = 0; e < 8; ++e) acc[e] = 0.0f;

    for (int d = lane; d < DDIM; d += 32) {
        float xv = xr[d];
        const float4* wr = (const float4*)(Wg + (size_t)d * NEXP);
        float4 w0 = wr[0], w1 = wr[1];
        acc[0] += xv * w0.x; acc[1] += xv * w0.y;
        acc[2] += xv * w0.z; acc[3] += xv * w0.w;
        acc[4] += xv * w1.x; acc[5] += xv * w1.y;
        acc[6] += xv * w1.z; acc[7] += xv * w1.w;
    }

    // wave32 butterfly reduction
#pragma unroll
    for (int off = 16; off > 0; off >>= 1) {
#pragma unroll
        for (int e = 0; e < 8; ++e) acc[e] += __shfl_xor(acc[e], off, 32);
    }

    if (lane == 0) {
        float v0 = -3.4e38f, v1 = -3.4e38f;
        int   i0 = 0,        i1 = 0;
#pragma unroll
        for (int e = 0; e < 8; ++e) {
            float v = acc[e] + bg[e];
            if (v > v0)      { v1 = v0; i1 = i0; v0 = v; i0 = e; }
            else if (v > v1) { v1 = v;  i1 = e; }
        }
        topIdx[token * 2 + 0] = i0; topIdx[token * 2 + 1] = i1;
        topW [token * 2 + 0] = v0; topW [token * 2 + 1] = v1;
        atomicAdd(&cnt[i0], 1);
        atomicAdd(&cnt[i1], 1);
    }
}

// ---------------------------------------------------------------------------
// Kernel 3: tiny scan -- pad counts to RT, tile->expert map, scatter cursors
// ---------------------------------------------------------------------------
__global__ void moe_scan(const int* __restrict__ cnt, int* __restrict__ cursor,
                         int* __restrict__ tileExpert)
{
    if (threadIdx.x != 0 || blockIdx.x != 0) return;
    for (int t = 0; t < MAXTILES; ++t) tileExpert[t] = -1;
    int off = 0;
    for (int e = 0; e < NEXP; ++e) {
        cursor[e] = off;
        int tiles = (cnt[e] + (RT - 1)) / RT;
        int tbase = off / RT;
        for (int t = 0; t < tiles; ++t) tileExpert[tbase + t] = e;
        off += tiles * RT;
    }
}

// ---------------------------------------------------------------------------
// Kernel 4: scatter token-expert pairs into contiguous per-expert segments
// ---------------------------------------------------------------------------
__global__ __launch_bounds__(256)
void moe_scatter(const int* __restrict__ topIdx, const float* __restrict__ topW,
                 int* __restrict__ cursor, int* __restrict__ pairTok,
                 float* __restrict__ pairW)
{
    int token = blockIdx.x * 256 + threadIdx.x;
    if (token >= NTOK) return;
#pragma unroll
    for (int k = 0; k < 2; ++k) {
        int   e = topIdx[token * 2 + k];
        float w = topW [token * 2 + k];
        int slot = atomicAdd(&cursor[e], 1);
        pairTok[slot] = token;
        pairW [slot] = w;
    }
}

// ---------------------------------------------------------------------------
// Kernel 5: grouped GEMM, 32 routed rows x 256 cols per block.
//   - x tile staged via CDNA5 async direct-to-LDS (global_load_async_to_lds),
//     tracked by ASYNCcnt; each thread owns one fixed row so token lookup /
//     clamp / base addresses hoist out of the issue loop.
//   - V_WMMA_F32_16X16X4_F32; each B fragment feeds both row-halves.
//   grid = (MAXTILES, DDIM/256), block = 128 (4 waves), wave w owns cols w*64..+63
// ---------------------------------------------------------------------------
__global__ __launch_bounds__(128)
void moe_gemm(const float* __restrict__ x, const float* __restrict__ We,
              const float* __restrict__ be,
              const int* __restrict__ tileExpert,
              const int* __restrict__ pairTok, const float* __restrict__ pairW,
              float* __restrict__ out)
{
    const int tile = blockIdx.x;
    const int e = tileExpert[tile];
    if (e < 0) return;                       // unused padded tile (block-uniform)

    const int lane   = threadIdx.x & 31;
    const int wave   = threadIdx.x >> 5;
    const int nlo    = lane & 15;
    const int half   = lane >> 4;            // 0: lanes 0-15, 1: lanes 16-31
    const int rowOff = half << 1;            // K-offset per ISA f32 WMMA layout

    __shared__ float xs[RT][KC + 4];         // +4 pad -> LDS bank stride 4
    __shared__ int   toks[RT];
    __shared__ float twt[RT];

    if (threadIdx.x < RT) {
        toks[threadIdx.x] = pairTok[tile * RT + threadIdx.x];
        twt [threadIdx.x] = pairW [tile * RT + threadIdx.x];
    }
    __syncthreads();

    // --- staging assignment: 4 threads per row, fixed row per thread ---
    const int rowS = threadIdx.x >> 2;       // 0..31
    const int cS   = threadIdx.x & 3;        // 0..3, 16B sub-column
    int tkS = toks[rowS];
    tkS = tkS < 0 ? 0 : tkS;                 // clamp padding rows (epilogue skips them)
    const unsigned long long gRowBase =
        (unsigned long long)(size_t)(x + (size_t)tkS * DDIM) + (unsigned)(cS * 16);
    const unsigned ldsRowBase = (unsigned)(size_t)&xs[rowS][0] + (unsigned)(cS * 16);

    const int colBase = blockIdx.y * 256 + wave * 64;
    const float* Bb = We + (size_t)e * DDIM * DDIM + colBase + nlo;

    const v8f vzero = {0.f, 0.f, 0.f, 0.f, 0.f, 0.f, 0.f, 0.f};
    v8f accL[4], accH[4];
#pragma unroll
    for (int t = 0; t < 4; ++t) { accL[t] = vzero; accH[t] = vzero; }

    for (int kb = 0; kb < DDIM; kb += KC) {
        __syncthreads();
        // Async-stage 32 rows x KC floats (32 KB) directly into LDS:
        // per thread 16 x b128, 64B stride, all addressing hoisted.
        unsigned long long ga = gRowBase + (unsigned)(kb * 4);
        unsigned ld = ldsRowBase;
#pragma unroll
        for (int j = 0; j < (KC * 4) / (4 * 16); ++j) {      // 16 issues
            asm volatile("global_load_async_to_lds_b128 %0, %1, off"
                         :: "v"(ld), "v"(ga) : "memory");
            ga += 64;
            ld += 64;
        }
        asm volatile("s_wait_asynccnt 0x0" ::: "memory");
        __syncthreads();

#pragma unroll 2
        for (int kk = 0; kk < KC; kk += 4) {
            // A 16x4 f32 fragments for both row-halves:
            //   lane m<16 -> K=kk,kk+1 ; lane>=16 -> K=kk+2,kk+3
            v2f aL, aH;
            aL.x = xs[nlo][kk + rowOff];
            aL.y = xs[nlo][kk + rowOff + 1];
            aH.x = xs[16 + nlo][kk + rowOff];
            aH.y = xs[16 + nlo][kk + rowOff + 1];
            // B 4x16 fragments: vgpr0 = row K+rowOff, vgpr1 = row K+rowOff+1
            const float* bp = Bb + (size_t)(kb + kk + rowOff) * DDIM;
            v2f b0, b1, b2, b3;
            b0.x = bp[0];   b0.y = bp[DDIM];
            b1.x = bp[16];  b1.y = bp[DDIM + 16];
            b2.x = bp[32];  b2.y = bp[DDIM + 32];
            b3.x = bp[48];  b3.y = bp[DDIM + 48];
            accL[0] = __builtin_amdgcn_wmma_f32_16x16x4_f32(false, aL, false, b0, (short)0, accL[0], false, false);
            accH[0] = __builtin_amdgcn_wmma_f32_16x16x4_f32(false, aH, false, b0, (short)0, accH[0], false, false);
            accL[1] = __builtin_amdgcn_wmma_f32_16x16x4_f32(false, aL, false, b1, (short)0, accL[1], false, false);
            accH[1] = __builtin_amdgcn_wmma_f32_16x16x4_f32(false, aH, false, b1, (short)0, accH[1], false, false);
            accL[2] = __builtin_amdgcn_wmma_f32_16x16x4_f32(false, aL, false, b2, (short)0, accL[2], false, false);
            accH[2] = __builtin_amdgcn_wmma_f32_16x16x4_f32(false, aH, false, b2, (short)0, accH[2], false, false);
            accL[3] = __builtin_amdgcn_wmma_f32_16x16x4_f32(false, aL, false, b3, (short)0, accL[3], false, false);
            accH[3] = __builtin_amdgcn_wmma_f32_16x16x4_f32(false, aH, false, b3, (short)0, accH[3], false, false);
        }
    }

    // Epilogue: (acc + bias) * logit, scatter-add into out.
    // C/D layout: vgpr r -> row M=r (lanes 0-15) / M=r+8 (lanes 16-31)
    const float* bias = be + (size_t)e * DDIM;
#pragma unroll
    for (int t = 0; t < 4; ++t) {
        const int n  = colBase + t * 16 + nlo;
        const float bv = bias[n];
#pragma unroll
        for (int r = 0; r < 8; ++r) {
            const int prow = r + (half << 3);
            const int tokL = toks[prow];
            if (tokL >= 0) {
                atomicAdd(out + (size_t)tokL * DDIM + n, (accL[t][r] + bv) * twt[prow]);
            }
            const int tokH = toks[16 + prow];
            if (tokH >= 0) {
                atomicAdd(out + (size_t)tokH * DDIM + n, (accH[t][r] + bv) * twt[16 + prow]);
            }
        }
    }
}

// ---------------------------------------------------------------------------
extern "C" void kernel_launch(void* const* d_in, const int* in_sizes, int n_in,
                              void* d_out, int out_size, void* d_ws, size_t ws_size,
                              hipStream_t stream)
{
    // setup_inputs order: x, router_mask(unused), We, be, Wg, bg, top_k(=2)
    const float* x  = (const float*)d_in[0];
    const float* We = (const float*)d_in[2];
    const float* be = (const float*)d_in[3];
    const float* Wg = (const float*)d_in[4];
    const float* bg = (const float*)d_in[5];
    float* out = (float*)d_out;

    // workspace layout (~530 KB)
    int*   cnt        = (int*)d_ws;                   // 8
    int*   cursor     = cnt + 8;                      // 8
    int*   tileExpert = cursor + 8;                   // MAXTILES
    int*   pairTok    = tileExpert + MAXTILES;        // PAIRCAP
    float* pairW      = (float*)(pairTok + PAIRCAP);  // PAIRCAP
    int*   topIdx     = (int*)(pairW + PAIRCAP);      // 2*NTOK
    float* topW       = (float*)(topIdx + 2 * NTOK);  // 2*NTOK

    hipLaunchKernelGGL(moe_init, dim3(1024), dim3(256), 0, stream,
                       out, cnt, pairTok, pairW);
    hipLaunchKernelGGL(moe_gate, dim3(NTOK / 8), dim3(256), 0, stream,
                       x, Wg, bg, topIdx, topW, cnt);
    hipLaunchKernelGGL(moe_scan, dim3(1), dim3(64), 0, stream,
                       cnt, cursor, tileExpert);
    hipLaunchKernelGGL(moe_scatter, dim3(NTOK / 256), dim3(256), 0, stream,
                       topIdx, topW, cursor, pairTok, pairW);
    hipLaunchKernelGGL(moe_gemm, dim3(MAXTILES, DDIM / 256), dim3(128), 0, stream,
                       x, We, be, tileExpert, pairTok, pairW, out);
}